// ConditionalFPS_74234214744566
// MI455X (gfx1250) — compile-verified
//
#include <hip/hip_runtime.h>
#include <math.h>

// Problem constants (match setup_inputs)
#define Bq   2
#define Nq   8192
#define Sq   2048
#define KNN  10
#define BN   (Bq * Nq)
#define PI_F       3.14159265358979f
#define HALF_PI_F  1.57079632679490f

typedef __attribute__((ext_vector_type(2))) float v2f;
typedef __attribute__((ext_vector_type(8))) float v8f;

// ---------------------------------------------------------------------------
// |p|^2 per point, and zero the FPS one-hot mask.
// ---------------------------------------------------------------------------
__global__ __launch_bounds__(256) void sq_kernel(const float* __restrict__ pos,
                                                 float* __restrict__ sq,
                                                 float* __restrict__ mask) {
  int i = blockIdx.x * blockDim.x + threadIdx.x;
  if (i < BN) {
    float x = pos[3 * i], y = pos[3 * i + 1], z = pos[3 * i + 2];
    sq[i] = x * x + y * y + z * z;
    mask[i] = 0.0f;
  }
}

// ---------------------------------------------------------------------------
// KNN via V_WMMA_F32_16X16X4_F32. One wave per (batch, 16-query tile).
// A = candidate tile (16x4, xyz0), B = query tile (4x16, xyz0).
// D lane layout: lane n (0..15) holds column N=n (query), rows M=v (cand);
// lanes 16..31 hold rows M=v+8 of the same query column. Each lane keeps a
// top-10 (score = 2*dot - |c|^2) fully in registers. Candidate loop unrolled
// x4 (4 independent WMMAs cover the WMMA->VALU hazard window); insertion
// network is skipped with a wave-uniform __any() threshold test (no EXEC
// change, WMMA-legal). Lane pairs merge via __shfl_xor(16).
// Emits knn_idx, maxd, dist_loss.
// ---------------------------------------------------------------------------
__global__ __launch_bounds__(32) void knn_kernel(const float* __restrict__ pos,
                                                 const float* __restrict__ sq,
                                                 int*   __restrict__ knn_idx,
                                                 float* __restrict__ maxd,
                                                 float* __restrict__ dist_loss) {
  const int lane  = threadIdx.x;
  const int b     = blockIdx.x / (Nq / 16);
  const int qtile = blockIdx.x % (Nq / 16);
  const int qbase = qtile * 16;
  const int col   = lane & 15;
  const bool hi   = lane >= 16;
  const int moff  = hi ? 8 : 0;

  const float* pb  = pos + (size_t)b * Nq * 3;
  const float* sqb = sq + (size_t)b * Nq;

  // B operand: query tile. VGPR0 = (K0 low | K2 high), VGPR1 = (K1 low | K3=0 high).
  const float* pq = pb + (size_t)(qbase + col) * 3;
  float qx = pq[0], qy = pq[1], qz = pq[2];
  v2f Bv;
  Bv.x = hi ? qz : qx;
  Bv.y = hi ? 0.0f : qy;
  const float sqq = sqb[qbase + col];

  float best[KNN];
  int   bidx[KNN];
#pragma unroll
  for (int j = 0; j < KNN; ++j) { best[j] = -3.0e38f; bidx[j] = 0; }

  constexpr int UNROLL = 4;
  for (int ct = 0; ct < Nq / 16; ct += UNROLL) {
    v2f Av[UNROLL];
#pragma unroll
    for (int u = 0; u < UNROLL; ++u) {
      const float* pc = pb + (size_t)((ct + u) * 16 + col) * 3;
      float cx = pc[0], cy = pc[1], cz = pc[2];
      Av[u].x = hi ? cz : cx;
      Av[u].y = hi ? 0.0f : cy;
      if (ct + UNROLL < Nq / 16)
        __builtin_prefetch(pb + (size_t)((ct + UNROLL + u) * 16 + col) * 3, 0, 0);
    }

    v8f acc[UNROLL];
#pragma unroll
    for (int u = 0; u < UNROLL; ++u) {
      v8f z = {};
      acc[u] = __builtin_amdgcn_wmma_f32_16x16x4_f32(
          /*neg_a=*/false, Av[u], /*neg_b=*/false, Bv,
          /*c_mod=*/(short)0, z, /*reuse_a=*/false, /*reuse_b=*/false);
    }

#pragma unroll
    for (int u = 0; u < UNROLL; ++u) {
      const int cbase = (ct + u) * 16;
      // |c|^2 for the 8 candidate rows owned by this lane (32B-aligned).
      const float4* sp = (const float4*)(sqb + cbase + moff);
      float4 s0 = sp[0], s1 = sp[1];
      float sc[8] = {s0.x, s0.y, s0.z, s0.w, s1.x, s1.y, s1.z, s1.w};

#pragma unroll
      for (int v = 0; v < 8; ++v) {
        float scr  = 2.0f * acc[u][v] - sc[v];   // monotone in -d2
        int   cidx = cbase + moff + v;
        // Wave-uniform early reject: skip insertion when no lane beats its
        // current 10th best. Uniform branch -> EXEC unchanged.
        if (__any(scr > best[KNN - 1])) {
          float cs = scr; int ci = cidx;
#pragma unroll
          for (int j = 0; j < KNN; ++j) {      // branch-free insertion network
            bool  sw = cs > best[j];
            float tb = best[j]; int ti = bidx[j];
            best[j] = sw ? cs : tb;  bidx[j] = sw ? ci : ti;
            cs      = sw ? tb : cs;  ci       = sw ? ti : ci;
          }
        }
      }
    }
  }

  // Merge the two half-lists (lanes n and n+16 share query column n).
#pragma unroll
  for (int j = 0; j < KNN; ++j) {
    float ob = __shfl_xor(best[j], 16, 32);
    int   oi = __shfl_xor(bidx[j], 16, 32);
    float cs = ob; int ci = oi;
#pragma unroll
    for (int t = 0; t < KNN; ++t) {
      bool  sw = cs > best[t];
      float tb = best[t]; int ti = bidx[t];
      best[t] = sw ? cs : tb;  bidx[t] = sw ? ci : ti;
      cs      = sw ? tb : cs;  ci       = sw ? ti : ci;
    }
  }

  if (!hi) {
    const int n = qbase + col;
    int* op = knn_idx + ((size_t)b * Nq + n) * KNN;
    float dsum = 0.0f, dmax = 0.0f;
#pragma unroll
    for (int j = 0; j < KNN; ++j) {
      op[j] = bidx[j];
      float d2 = sqq - best[j];
      d2 = d2 > 0.0f ? d2 : 0.0f;
      float d = sqrtf(d2);
      dsum += d;
      dmax = d > dmax ? d : dmax;
    }
    maxd[(size_t)b * Nq + n]      = dmax;
    dist_loss[(size_t)b * Nq + n] = dmax + dsum * (1.0f / KNN);
  }
}

// ---------------------------------------------------------------------------
// Furthest point sampling: one workgroup per batch. SoA positions AND dmin
// live in (dynamic) LDS -- ~140KB of the WGP's 320KB -- so the 2047
// sequential argmax sweeps generate zero global traffic.
// ---------------------------------------------------------------------------
__global__ __launch_bounds__(1024) void fps_kernel(const float* __restrict__ pos,
                                                   int*   __restrict__ p_idx,
                                                   float* __restrict__ mask) {
  extern __shared__ float lds[];
  float* sx   = lds;              // Nq
  float* sy   = sx + Nq;          // Nq
  float* sz   = sy + Nq;          // Nq
  float* dmin = sz + Nq;          // Nq
  float* rv   = dmin + Nq;        // 1024
  int*   ri   = (int*)(rv + 1024);// 1024
  __shared__ int s_last;

  const int b = blockIdx.x, tid = threadIdx.x;
  const float* pb = pos + (size_t)b * Nq * 3;

  for (int n = tid; n < Nq; n += 1024) {
    sx[n] = pb[3 * n];
    sy[n] = pb[3 * n + 1];
    sz[n] = pb[3 * n + 2];
    dmin[n] = 3.4e38f;
  }
  if (tid == 0) {
    s_last = 0;
    p_idx[b * Sq] = 0;
    mask[(size_t)b * Nq] = 1.0f;
  }
  __syncthreads();

  for (int it = 1; it < Sq; ++it) {
    const int last = s_last;
    const float lx = sx[last], ly = sy[last], lz = sz[last];
    float lb = -3.4e38f; int li = 0;
    for (int n = tid; n < Nq; n += 1024) {
      float dx = sx[n] - lx, dy = sy[n] - ly, dz = sz[n] - lz;
      float d = dx * dx + dy * dy + dz * dz;
      float m = dmin[n];
      m = d < m ? d : m;
      dmin[n] = m;
      if (m > lb) { lb = m; li = n; }
    }
    rv[tid] = lb; ri[tid] = li;
    __syncthreads();
    for (int off = 512; off > 0; off >>= 1) {
      if (tid < off) {
        float ov = rv[tid + off]; int oi = ri[tid + off];
        if (ov > rv[tid] || (ov == rv[tid] && oi < ri[tid])) { rv[tid] = ov; ri[tid] = oi; }
      }
      __syncthreads();
    }
    if (tid == 0) {
      int sel = ri[0];
      s_last = sel;
      p_idx[b * Sq + it] = sel;
      mask[(size_t)b * Nq + sel] = 1.0f;
    }
    __syncthreads();
  }
}

// ---------------------------------------------------------------------------
// Per-point curvature (angle sum vs normal) and raw density K/maxd^3.
// ---------------------------------------------------------------------------
__global__ __launch_bounds__(256) void feat_kernel(const float* __restrict__ pos,
                                                   const float* __restrict__ x,
                                                   const int*   __restrict__ knn_idx,
                                                   const float* __restrict__ maxd,
                                                   float* __restrict__ angle,
                                                   float* __restrict__ dense) {
  int i = blockIdx.x * blockDim.x + threadIdx.x;
  if (i >= BN) return;
  const int b = i / Nq, n = i % Nq;
  const float* pb = pos + (size_t)b * Nq * 3;
  const float nx = x[((size_t)b * 3 + 0) * Nq + n];
  const float ny = x[((size_t)b * 3 + 1) * Nq + n];
  const float nz = x[((size_t)b * 3 + 2) * Nq + n];
  const int* id = knn_idx + (size_t)i * KNN;
  float asum = 0.0f;
#pragma unroll
  for (int j = 0; j < KNN; ++j) {
    int t = id[j];
    float inner = pb[3 * t] * nx + pb[3 * t + 1] * ny + pb[3 * t + 2] * nz;
    inner = fminf(1.0f, fmaxf(-1.0f, inner));
    float a = acosf(inner);
    a = a > HALF_PI_F ? (PI_F - a) : a;
    asum += a;
  }
  angle[i] = asum;
  float md = maxd[i];
  dense[i] = (md > 0.0f) ? ((float)KNN / (md * md * md)) : __builtin_inff();
}

// ---------------------------------------------------------------------------
// Global stats: angle sum, finite max of dense, dense sum with inf replaced.
// ---------------------------------------------------------------------------
__global__ __launch_bounds__(1024) void stats_kernel(const float* __restrict__ angle,
                                                     const float* __restrict__ dense,
                                                     float* __restrict__ stats) {
  __shared__ float r1[1024];
  __shared__ float r2[1024];
  __shared__ float s_as, s_dm;
  const int tid = threadIdx.x;

  float as = 0.0f, dm = -3.4e38f;
  for (int i = tid; i < BN; i += 1024) {
    as += angle[i];
    float de = dense[i];
    if (!__builtin_isinf(de) && de > dm) dm = de;
  }
  r1[tid] = as; r2[tid] = dm;
  __syncthreads();
  for (int off = 512; off > 0; off >>= 1) {
    if (tid < off) {
      r1[tid] += r1[tid + off];
      r2[tid] = fmaxf(r2[tid], r2[tid + off]);
    }
    __syncthreads();
  }
  if (tid == 0) { s_as = r1[0]; s_dm = r2[0]; }
  __syncthreads();

  const float dmaxv = s_dm;
  float ds = 0.0f;
  for (int i = tid; i < BN; i += 1024) {
    float de = dense[i];
    if (__builtin_isinf(de)) de = dmaxv;
    ds += de;
  }
  r1[tid] = ds;
  __syncthreads();
  for (int off = 512; off > 0; off >>= 1) {
    if (tid < off) r1[tid] += r1[tid + off];
    __syncthreads();
  }
  if (tid == 0) { stats[0] = s_as; stats[1] = dmaxv; stats[2] = r1[0]; }
}

// ---------------------------------------------------------------------------
// opt = feats . w + b, then per-batch softmax (one block per batch).
// ---------------------------------------------------------------------------
__global__ __launch_bounds__(1024) void score_kernel(const float* __restrict__ mask,
                                                     const float* __restrict__ angle,
                                                     const float* __restrict__ dense,
                                                     const float* __restrict__ stats,
                                                     const float* __restrict__ w,
                                                     const float* __restrict__ bias,
                                                     float* __restrict__ smax) {
  __shared__ float red[1024];
  __shared__ float s_red;
  const int b = blockIdx.x, tid = threadIdx.x;
  const float w0 = w[0], w1 = w[1], w2 = w[2], b0 = bias[0];
  const float asum = stats[0], amean = stats[0] / (float)BN;
  const float dmax = stats[1], dsum = stats[2], dmean = stats[2] / (float)BN;
  const float fmean = (float)Sq / (float)Nq;           // (B*S)/(B*N)
  const float fsum  = (float)(Bq * Sq);
  const size_t base = (size_t)b * Nq;

  float lmax = -3.4e38f;
  for (int n = tid; n < Nq; n += 1024) {
    float f = (mask[base + n] - fmean) / fsum;
    float c = (angle[base + n] - amean) / asum;
    float de = dense[base + n];
    if (__builtin_isinf(de)) de = dmax;
    float dn = (de - dmean) / dsum;
    float o = f * w0 + c * w1 + dn * w2 + b0;
    smax[base + n] = o;
    lmax = o > lmax ? o : lmax;
  }
  red[tid] = lmax;
  __syncthreads();
  for (int off = 512; off > 0; off >>= 1) {
    if (tid < off) red[tid] = fmaxf(red[tid], red[tid + off]);
    __syncthreads();
  }
  if (tid == 0) s_red = red[0];
  __syncthreads();
  const float mx = s_red;

  float lsum = 0.0f;
  for (int n = tid; n < Nq; n += 1024) {
    float e = expf(smax[base + n] - mx);
    smax[base + n] = e;
    lsum += e;
  }
  __syncthreads();
  red[tid] = lsum;
  __syncthreads();
  for (int off = 512; off > 0; off >>= 1) {
    if (tid < off) red[tid] += red[tid + off];
    __syncthreads();
  }
  if (tid == 0) s_red = red[0];
  __syncthreads();
  const float inv = 1.0f / s_red;
  for (int n = tid; n < Nq; n += 1024) smax[base + n] *= inv;
}

// ---------------------------------------------------------------------------
// Top-2048 of smax per batch (sequential argmax extraction, keys in LDS).
// Writes int indices to ws and float indices to d_out[0 .. B*S).
// ---------------------------------------------------------------------------
__global__ __launch_bounds__(1024) void topk_kernel(const float* __restrict__ smax,
                                                    int*   __restrict__ top_idx,
                                                    float* __restrict__ out) {
  __shared__ float key[Nq];
  __shared__ float rv[1024];
  __shared__ int   ri[1024];
  const int b = blockIdx.x, tid = threadIdx.x;
  const float* s = smax + (size_t)b * Nq;
  for (int n = tid; n < Nq; n += 1024) key[n] = s[n];
  __syncthreads();

  for (int it = 0; it < Sq; ++it) {
    float lb = -3.4e38f; int li = 0;
    for (int n = tid; n < Nq; n += 1024) {
      float v = key[n];
      if (v > lb) { lb = v; li = n; }
    }
    rv[tid] = lb; ri[tid] = li;
    __syncthreads();
    for (int off = 512; off > 0; off >>= 1) {
      if (tid < off) {
        float ov = rv[tid + off]; int oi = ri[tid + off];
        if (ov > rv[tid] || (ov == rv[tid] && oi < ri[tid])) { rv[tid] = ov; ri[tid] = oi; }
      }
      __syncthreads();
    }
    if (tid == 0) {
      int sel = ri[0];
      top_idx[b * Sq + it] = sel;
      out[b * Sq + it] = (float)sel;
      key[sel] = -3.4e38f;
    }
    __syncthreads();
  }
}

// ---------------------------------------------------------------------------
// Losses. sd_loss == dist_loss gathered at top_idx; bdist == at p_idx.
// losses = [L0, L0, L2, L3] appended after the B*S index outputs.
// ---------------------------------------------------------------------------
__global__ __launch_bounds__(1024) void loss_kernel(const float* __restrict__ dl,
                                                    const float* __restrict__ smax,
                                                    const int*   __restrict__ top_idx,
                                                    const int*   __restrict__ p_idx,
                                                    float* __restrict__ out) {
  __shared__ float r[1024];
  const int tid = threadIdx.x;

  float a = 0.0f;
  for (int i = tid; i < BN; i += 1024) a += dl[i] * smax[i];
  r[tid] = a;
  __syncthreads();
  for (int off = 512; off > 0; off >>= 1) { if (tid < off) r[tid] += r[tid + off]; __syncthreads(); }
  const float L0 = r[0] / (float)BN;
  __syncthreads();

  float c = 0.0f;
  for (int i = tid; i < Bq * Sq; i += 1024) {
    int b = i / Sq;
    c += dl[(size_t)b * Nq + top_idx[i]];
  }
  r[tid] = c;
  __syncthreads();
  for (int off = 512; off > 0; off >>= 1) { if (tid < off) r[tid] += r[tid + off]; __syncthreads(); }
  const float L2 = r[0] / (float)(Bq * Sq);
  __syncthreads();

  float d = 0.0f;
  for (int i = tid; i < Bq * Sq; i += 1024) {
    int b = i / Sq;
    d += dl[(size_t)b * Nq + p_idx[i]];
  }
  r[tid] = d;
  __syncthreads();
  for (int off = 512; off > 0; off >>= 1) { if (tid < off) r[tid] += r[tid + off]; __syncthreads(); }
  const float L3 = r[0] / (float)(Bq * Sq);

  if (tid == 0) {
    out[Bq * Sq + 0] = L0;
    out[Bq * Sq + 1] = L0;
    out[Bq * Sq + 2] = L2;
    out[Bq * Sq + 3] = L3;
  }
}

// ---------------------------------------------------------------------------
extern "C" void kernel_launch(void* const* d_in, const int* in_sizes, int n_in,
                              void* d_out, int out_size, void* d_ws, size_t ws_size,
                              hipStream_t stream) {
  (void)in_sizes; (void)n_in; (void)out_size; (void)ws_size;
  const float* x   = (const float*)d_in[0];   // (B,3,N)
  const float* pos = (const float*)d_in[1];   // (B,N,3)
  const float* sw  = (const float*)d_in[2];   // (1,3)
  const float* sb  = (const float*)d_in[3];   // (1,)
  float* out = (float*)d_out;

  float* wsf   = (float*)d_ws;
  float* sq    = wsf;                 // BN
  float* mask  = sq + BN;             // BN
  float* maxd  = mask + BN;           // BN
  float* dl    = maxd + BN;           // BN
  float* angle = dl + BN;             // BN
  float* dense = angle + BN;          // BN
  float* smax  = dense + BN;          // BN
  float* stats = smax + BN;           // 16
  int* knn_idx = (int*)(stats + 16);            // BN*KNN
  int* p_idx   = knn_idx + (size_t)BN * KNN;    // B*S
  int* top_idx = p_idx + Bq * Sq;               // B*S

  const size_t fps_lds = (size_t)(4 * Nq + 2 * 1024) * sizeof(float);

  sq_kernel  <<<(BN + 255) / 256, 256, 0, stream>>>(pos, sq, mask);
  knn_kernel <<<Bq * (Nq / 16), 32, 0, stream>>>(pos, sq, knn_idx, maxd, dl);
  fps_kernel <<<Bq, 1024, fps_lds, stream>>>(pos, p_idx, mask);
  feat_kernel<<<(BN + 255) / 256, 256, 0, stream>>>(pos, x, knn_idx, maxd, angle, dense);
  stats_kernel<<<1, 1024, 0, stream>>>(angle, dense, stats);
  score_kernel<<<Bq, 1024, 0, stream>>>(mask, angle, dense, stats, sw, sb, smax);
  topk_kernel <<<Bq, 1024, 0, stream>>>(smax, top_idx, out);
  loss_kernel <<<1, 1024, 0, stream>>>(dl, smax, top_idx, p_idx, out);
}